// VisionEncoderMambaBlock_66571993088871
// MI455X (gfx1250) — compile-verified
//
#include <hip/hip_runtime.h>
#include <math.h>

typedef float v2f __attribute__((ext_vector_type(2)));
typedef float v8f __attribute__((ext_vector_type(8)));

#define DIMD   256
#define DSTATE 16
#define DTRANK 16
#define BATCH  4
#define SEQ    4096

__device__ __forceinline__ float softplus_f(float v) {
  return (v > 20.f) ? v : log1pf(__expf(v));
}
__device__ __forceinline__ float silu_f(float v) {
  return v / (1.f + __expf(-v));
}

// ---------------- LayerNorm: one wave per 256-wide row ----------------
__global__ void k_layernorm(const float* __restrict__ x, const float* __restrict__ g,
                            const float* __restrict__ b, float* __restrict__ xn, int rows) {
  int wid  = (blockIdx.x * blockDim.x + threadIdx.x) >> 5;
  int lane = threadIdx.x & 31;
  if (wid >= rows) return;
  const float4* r4 = (const float4*)(x + (size_t)wid * DIMD + lane * 8);
  float4 va = r4[0], vb = r4[1];
  float s  = va.x + va.y + va.z + va.w + vb.x + vb.y + vb.z + vb.w;
  float s2 = va.x*va.x + va.y*va.y + va.z*va.z + va.w*va.w +
             vb.x*vb.x + vb.y*vb.y + vb.z*vb.z + vb.w*vb.w;
#pragma unroll
  for (int m = 1; m < 32; m <<= 1) {
    s  += __shfl_xor(s,  m, 32);
    s2 += __shfl_xor(s2, m, 32);
  }
  float mu  = s  * (1.f / DIMD);
  float var = s2 * (1.f / DIMD) - mu * mu;
  float rs  = rsqrtf(var + 1e-5f);
  const float4* g4 = (const float4*)(g + lane * 8);
  const float4* b4 = (const float4*)(b + lane * 8);
  float4 ga = g4[0], gb = g4[1], ba = b4[0], bb = b4[1];
  float4 oa, ob;
  oa.x = (va.x - mu) * rs * ga.x + ba.x;  oa.y = (va.y - mu) * rs * ga.y + ba.y;
  oa.z = (va.z - mu) * rs * ga.z + ba.z;  oa.w = (va.w - mu) * rs * ga.w + ba.w;
  ob.x = (vb.x - mu) * rs * gb.x + bb.x;  ob.y = (vb.y - mu) * rs * gb.y + bb.y;
  ob.z = (vb.z - mu) * rs * gb.z + bb.z;  ob.w = (vb.w - mu) * rs * gb.w + bb.w;
  float4* o4 = (float4*)(xn + (size_t)wid * DIMD + lane * 8);
  o4[0] = oa; o4[1] = ob;
}

// ---------------- fp32 WMMA GEMM: C = act(A[M,KD] @ Bop + bias) ----------------
// All shape/stride params are compile-time: loads become base + imm-offset
// (24-bit signed offsets cover the whole panel), no per-iteration pointer math.
// TRANSW==0: W is [KD,N] row-major. TRANSW==1: W is [N,KD]; computes A @ W^T.
// One wave computes a 16 x (16*NT) strip; A fragment reused across NT accumulators.
// A-frag (ISA 16x4 f32): lanes 0-15 hold K=k0,k0+1; lanes 16-31 hold K=k0+2,k0+3.
// B-frag (4x16): lanes 0-15 rows k0,k0+1; lanes 16-31 rows k0+2,k0+3; col = lane&15.
// C/D: vgpr i -> row (half*8+i), col = lane&15.
template <int NT, int TRANSW, int ACT, int KD, int LDA, int LDW, int LDC>
__global__ void k_gemm(const float* __restrict__ A, const float* __restrict__ W,
                       const float* __restrict__ bias, float* __restrict__ C,
                       int M, int N) {
  int wid  = (blockIdx.x * blockDim.x + threadIdx.x) >> 5;
  int lane = threadIdx.x & 31;
  int ntn   = N / (16 * NT);
  int tiles = (M >> 4) * ntn;
  if (wid >= tiles) return;            // wave-uniform: EXEC stays all-ones for WMMA
  int tm = wid / ntn;
  int tn = (wid % ntn) * NT;
  int half = lane >> 4;
  int l    = lane & 15;
  int col0 = (tn << 4) + l;

  v8f acc[NT];
#pragma unroll
  for (int j = 0; j < NT; ++j) {
    float bv = bias ? bias[col0 + 16 * j] : 0.f;
#pragma unroll
    for (int i = 0; i < 8; ++i) acc[j][i] = bv;
  }

  const float* ap = A + (size_t)((tm << 4) + l) * LDA + 2 * half;
  const float* wp[NT];
#pragma unroll
  for (int j = 0; j < NT; ++j) {
    if (TRANSW) wp[j] = W + (size_t)(col0 + 16 * j) * LDW + 2 * half;
    else        wp[j] = W + (size_t)(2 * half) * LDW + col0 + 16 * j;
  }

#pragma unroll 8
  for (int k0 = 0; k0 < KD; k0 += 4) {
    v2f a = *(const v2f*)(ap + k0);
#pragma unroll
    for (int j = 0; j < NT; ++j) {
      v2f b;
      if (TRANSW) {
        b = *(const v2f*)(wp[j] + k0);
      } else {
        b.x = wp[j][k0 * LDW];
        b.y = wp[j][k0 * LDW + LDW];
      }
      acc[j] = __builtin_amdgcn_wmma_f32_16x16x4_f32(false, a, false, b, (short)0,
                                                     acc[j], false, false);
    }
  }

  int row0 = (tm << 4) + (half << 3);
#pragma unroll
  for (int j = 0; j < NT; ++j) {
#pragma unroll
    for (int i = 0; i < 8; ++i) {
      float vv = acc[j][i];
      if (ACT == 1) vv = softplus_f(vv);
      C[(size_t)(row0 + i) * LDC + col0 + 16 * j] = vv;
    }
  }
}

// ---------------- Fused bidirectional selective scan ----------------
// One launch covers BOTH directions (1024 concurrent waves; two separate kernel
// launches would be stream-serialized and halve latency hiding).
// One wave per 2 channels: lane = (half -> channel, n = lane&15 -> state).
// Pairwise chain compression: critical path is ONE v_fma per TWO timesteps:
//   h_{t+1} = (dA1*dA0)*h + (dA1*bx0 + bx1);  h_t = dA0*h + bx0 (side branch).
// C-projection fused: the [b,s,d,n] state tensor is never materialized.
__global__ void k_scan2(const float* __restrict__ xF, const float* __restrict__ dF,
                        const float* __restrict__ dbcF, const float* __restrict__ AlogF,
                        const float* __restrict__ DF, float* __restrict__ yF,
                        const float* __restrict__ xR, const float* __restrict__ dR,
                        const float* __restrict__ dbcR, const float* __restrict__ AlogR,
                        const float* __restrict__ DR, float* __restrict__ yR) {
  const int NBLK = BATCH * (DIMD / 2);              // 512 per direction
  int blk = blockIdx.x;
  int rev = (blk >= NBLK) ? 1 : 0;
  if (rev) blk -= NBLK;

  const float* xin   = rev ? xR    : xF;
  const float* delta = rev ? dR    : dF;
  const float* dbc   = rev ? dbcR  : dbcF;
  const float* Alog  = rev ? AlogR : AlogF;
  const float* Dp    = rev ? DR    : DF;
  float*       y     = rev ? yR    : yF;

  int lane = threadIdx.x;
  int half = lane >> 4, n = lane & 15;
  int bidx = blk >> 7;                              // 128 channel-pairs per batch
  int c    = ((blk & 127) << 1) + half;

  float Aln = -__expf(Alog[(size_t)c * DSTATE + n]);
  float Dd  = Dp[c];

  const float* xb = xin   + (size_t)bidx * SEQ * DIMD + c;
  const float* db = delta + (size_t)bidx * SEQ * DIMD + c;
  const float* Bp = dbc   + (size_t)bidx * SEQ * 48 + DTRANK + n;
  const float* Cp = Bp + DSTATE;
  float*       yb = y     + (size_t)bidx * SEQ * DIMD + c;

  float h = 0.f;
  int t   = rev ? (SEQ - 1) : 0;
  int stp = rev ? -1 : 1;
  for (int it = 0; it < SEQ; it += 2, t += 2 * stp) {
    int t0 = t, t1 = t + stp;
    int tp = t + stp * 16;
    if ((unsigned)tp < (unsigned)SEQ) {             // wave-uniform branch
      __builtin_prefetch(xb + (size_t)tp * DIMD, 0, 1);
      __builtin_prefetch(Bp + (size_t)tp * 48, 0, 1);
    }
    size_t a0 = (size_t)t0 * DIMD, a1 = (size_t)t1 * DIMD;
    size_t o0 = (size_t)t0 * 48,   o1 = (size_t)t1 * 48;
    float xv0 = xb[a0], dv0 = db[a0], Bv0 = Bp[o0], Cv0 = Cp[o0];
    float xv1 = xb[a1], dv1 = db[a1], Bv1 = Bp[o1], Cv1 = Cp[o1];
    float dA0 = __expf(dv0 * Aln);
    float dA1 = __expf(dv1 * Aln);
    float bx0 = dv0 * Bv0 * xv0;
    float bx1 = dv1 * Bv1 * xv1;
    float P   = dA1 * dA0;                          // off-chain
    float Q   = fmaf(dA1, bx0, bx1);                // off-chain
    float h0  = fmaf(dA0, h, bx0);                  // state at t0 (side branch)
    h = fmaf(P, h, Q);                              // state at t1 (critical path)
    float p0 = h0 * Cv0;
    float p1 = h * Cv1;
#pragma unroll
    for (int m = 1; m <= 8; m <<= 1) {
      p0 += __shfl_xor(p0, m, 32);
      p1 += __shfl_xor(p1, m, 32);
    }
    if (n == 0) {
      yb[a0] = fmaf(Dd, xv0, p0);
      yb[a1] = fmaf(Dd, xv1, p1);
    }
  }
}

// ---------------- Final combine: out = (silu(y1)+silu(y2))*silu(z1) + x ----------------
__global__ void k_combine(const float* __restrict__ x, const float* __restrict__ z1,
                          const float* __restrict__ y1, const float* __restrict__ y2,
                          float* __restrict__ out, int n4) {
  int i = blockIdx.x * blockDim.x + threadIdx.x;
  if (i >= n4) return;
  float4 xv = ((const float4*)x)[i];
  float4 zv = ((const float4*)z1)[i];
  float4 a  = ((const float4*)y1)[i];
  float4 b  = ((const float4*)y2)[i];
  float4 o;
  o.x = (silu_f(a.x) + silu_f(b.x)) * silu_f(zv.x) + xv.x;
  o.y = (silu_f(a.y) + silu_f(b.y)) * silu_f(zv.y) + xv.y;
  o.z = (silu_f(a.z) + silu_f(b.z)) * silu_f(zv.z) + xv.z;
  o.w = (silu_f(a.w) + silu_f(b.w)) * silu_f(zv.w) + xv.w;
  ((float4*)out)[i] = o;
}

extern "C" void kernel_launch(void* const* d_in, const int* in_sizes, int n_in,
                              void* d_out, int out_size, void* d_ws, size_t ws_size,
                              hipStream_t stream) {
  (void)in_sizes; (void)n_in; (void)out_size; (void)ws_size;
  const float* x      = (const float*)d_in[0];
  const float* ln_g   = (const float*)d_in[1];
  const float* ln_b   = (const float*)d_in[2];
  const float* Wp     = (const float*)d_in[3];
  const float* bp     = (const float*)d_in[4];
  const float* Wcf    = (const float*)d_in[5];
  const float* bcf    = (const float*)d_in[6];
  const float* Wcb    = (const float*)d_in[7];
  const float* bcb    = (const float*)d_in[8];
  const float* Wdbc_f = (const float*)d_in[9];
  const float* Wdt_f  = (const float*)d_in[10];
  const float* bdt_f  = (const float*)d_in[11];
  const float* Alog_f = (const float*)d_in[12];
  const float* D_f    = (const float*)d_in[13];
  const float* Wdbc_b = (const float*)d_in[14];
  const float* Wdt_b  = (const float*)d_in[15];
  const float* bdt_b  = (const float*)d_in[16];
  const float* Alog_b = (const float*)d_in[17];
  const float* D_b    = (const float*)d_in[18];
  float* out = (float*)d_out;

  const int    Mrows = BATCH * SEQ;             // 16384
  const size_t NEL   = (size_t)Mrows * DIMD;    // 4.19M floats
  const size_t NDBC  = (size_t)Mrows * 48;

  float* ws      = (float*)d_ws;
  float* xn      = ws;                // dead after z1 GEMM -> reused as y1
  float* z1      = ws + 1 * NEL;
  float* xf      = ws + 2 * NEL;
  float* xbck    = ws + 3 * NEL;
  float* delta_f = ws + 4 * NEL;
  float* delta_b = ws + 5 * NEL;
  float* y2      = ws + 6 * NEL;      // own buffer: scans run concurrently now
  float* dbc_f   = ws + 7 * NEL;
  float* dbc_b   = ws + 7 * NEL + NDBC;
  float* y1 = xn;

  // 1. LayerNorm (8 rows per 256-thread block)
  k_layernorm<<<(Mrows + 7) / 8, 256, 0, stream>>>(x, ln_g, ln_b, xn, Mrows);

  // 2. z1 = xn @ Wp + bp
  {
    int tiles = (Mrows / 16) * (DIMD / 64);
    k_gemm<4, 0, 0, 256, 256, 256, 256><<<(tiles + 7) / 8, 256, 0, stream>>>(
        xn, Wp, bp, z1, Mrows, DIMD);
  }
  // 3. conv(k=1): softplus(z1 @ Wc^T + bc)
  {
    int tiles = (Mrows / 16) * (DIMD / 64);
    k_gemm<4, 1, 1, 256, 256, 256, 256><<<(tiles + 7) / 8, 256, 0, stream>>>(
        z1, Wcf, bcf, xf, Mrows, DIMD);
    k_gemm<4, 1, 1, 256, 256, 256, 256><<<(tiles + 7) / 8, 256, 0, stream>>>(
        z1, Wcb, bcb, xbck, Mrows, DIMD);
  }
  // 4. dbc = x @ Wdbc -> [dt(16) | B(16) | C(16)]   (N=48: NT=3)
  {
    int tiles = (Mrows / 16);
    k_gemm<3, 0, 0, 256, 256, 48, 48><<<(tiles + 7) / 8, 256, 0, stream>>>(
        xf, Wdbc_f, nullptr, dbc_f, Mrows, 48);
    k_gemm<3, 0, 0, 256, 256, 48, 48><<<(tiles + 7) / 8, 256, 0, stream>>>(
        xbck, Wdbc_b, nullptr, dbc_b, Mrows, 48);
  }
  // 5. delta = softplus(dt_r @ Wdt + bdt)      (A = dbc[:, :16], lda=48, K=16)
  {
    int tiles = (Mrows / 16) * (DIMD / 64);
    k_gemm<4, 0, 1, 16, 48, 256, 256><<<(tiles + 7) / 8, 256, 0, stream>>>(
        dbc_f, Wdt_f, bdt_f, delta_f, Mrows, DIMD);
    k_gemm<4, 0, 1, 16, 48, 256, 256><<<(tiles + 7) / 8, 256, 0, stream>>>(
        dbc_b, Wdt_b, bdt_b, delta_b, Mrows, DIMD);
  }
  // 6. fused bidirectional selective scan (both directions in one launch)
  k_scan2<<<2 * BATCH * (DIMD / 2), 32, 0, stream>>>(
      xf, delta_f, dbc_f, Alog_f, D_f, y1,
      xbck, delta_b, dbc_b, Alog_b, D_b, y2);
  // 7. combine + residual
  int n4 = (int)(NEL / 4);
  k_combine<<<(n4 + 255) / 256, 256, 0, stream>>>(x, z1, y1, y2, out, n4);
}